// GTS_Spike_Decoding_38268158607753
// MI455X (gfx1250) — compile-verified
//
#include <hip/hip_runtime.h>
#include <math.h>

#define N_NODES 20000
#define N_EDGES 320000
#define EMB 256
#define DIN 512          // 2*EMB
#define LIN 100
#define L1 31
#define L2 8
#define C1 32
#define NKSTEP 640       // 5 blocks * 512 K / 4
#define BN_EPS 1e-5f

typedef float v2f __attribute__((ext_vector_type(2)));
typedef float v8f __attribute__((ext_vector_type(8)));

// ---------------- CNN encoder ----------------

__global__ void k_conv1(const float* __restrict__ x, const float* __restrict__ w,
                        const float* __restrict__ b, float* __restrict__ y,
                        float* __restrict__ stats) {
  __shared__ float sx[LIN];
  __shared__ float sw[C1 * 10];
  __shared__ float ssum[C1], ssq[C1];
  const int n = blockIdx.x, tid = threadIdx.x;
  if (tid < LIN) sx[tid] = x[n * LIN + tid];
  for (int i = tid; i < C1 * 10; i += blockDim.x) sw[i] = w[i];
  if (tid < C1) { ssum[tid] = 0.f; ssq[tid] = 0.f; }
  __syncthreads();
  for (int o = tid; o < C1 * L1; o += blockDim.x) {
    const int c = o / L1, l = o % L1;
    float acc = b[c];
#pragma unroll
    for (int t = 0; t < 10; ++t) acc += sw[c * 10 + t] * sx[3 * l + t];
    acc = fmaxf(acc, 0.f);
    y[(long)n * (C1 * L1) + o] = acc;
    atomicAdd(&ssum[c], acc);
    atomicAdd(&ssq[c], acc * acc);
  }
  __syncthreads();
  if (tid < C1) {
    atomicAdd(&stats[tid], ssum[tid]);
    atomicAdd(&stats[C1 + tid], ssq[tid]);
  }
}

__global__ void k_bn1(float* __restrict__ y, const float* __restrict__ stats,
                      const float* __restrict__ gamma, const float* __restrict__ beta) {
  const long i = (long)blockIdx.x * blockDim.x + threadIdx.x;
  if (i >= (long)N_NODES * C1 * L1) return;
  const int c = (int)((i / L1) % C1);
  const float cnt = (float)N_NODES * (float)L1;
  const float m = stats[c] / cnt;
  const float v = stats[C1 + c] / cnt - m * m;
  y[i] = (y[i] - m) * rsqrtf(v + BN_EPS) * gamma[c] + beta[c];
}

__global__ void k_conv2(const float* __restrict__ h, const float* __restrict__ w,
                        const float* __restrict__ b, float* __restrict__ y,
                        float* __restrict__ stats) {
  __shared__ float sh[C1 * L1];        // 992
  __shared__ float sw[C1 * C1 * 10];   // 10240
  __shared__ float ssum[C1], ssq[C1];
  const int n = blockIdx.x, tid = threadIdx.x;   // blockDim.x == 256
  for (int i = tid; i < C1 * L1; i += 256) sh[i] = h[(long)n * (C1 * L1) + i];
  for (int i = tid; i < C1 * C1 * 10; i += 256) sw[i] = w[i];
  if (tid < C1) { ssum[tid] = 0.f; ssq[tid] = 0.f; }
  __syncthreads();
  const int co = tid >> 3, l = tid & 7;
  float acc = b[co];
  for (int ci = 0; ci < C1; ++ci) {
    const float* hp = &sh[ci * L1 + 3 * l];
    const float* wp = &sw[(co * C1 + ci) * 10];
#pragma unroll
    for (int t = 0; t < 10; ++t) acc += wp[t] * hp[t];
  }
  acc = fmaxf(acc, 0.f);
  y[(long)n * EMB + tid] = acc;        // X layout [N,256], idx = co*8+l == tid
  atomicAdd(&ssum[co], acc);
  atomicAdd(&ssq[co], acc * acc);
  __syncthreads();
  if (tid < C1) {
    atomicAdd(&stats[tid], ssum[tid]);
    atomicAdd(&stats[C1 + tid], ssq[tid]);
  }
}

__global__ void k_bn2(const float* __restrict__ y, const float* __restrict__ stats,
                      const float* __restrict__ gamma, const float* __restrict__ beta,
                      const float* __restrict__ H0,
                      float* __restrict__ XH, float* __restrict__ XRH) {
  const long i = (long)blockIdx.x * blockDim.x + threadIdx.x;
  if (i >= (long)N_NODES * EMB) return;
  const int n = (int)(i / EMB), o = (int)(i % EMB);
  const int c = o >> 3;
  const float cnt = (float)N_NODES * (float)L2;
  const float m = stats[c] / cnt;
  const float v = stats[C1 + c] / cnt - m * m;
  const float xv = (y[i] - m) * rsqrtf(v + BN_EPS) * gamma[c] + beta[c];
  XH[(long)n * DIN + o] = xv;
  XRH[(long)n * DIN + o] = xv;
  XH[(long)n * DIN + EMB + o] = H0[i];
}

// ---------------- graph diffusion ----------------

__global__ void k_deg(const int* __restrict__ row, const int* __restrict__ col,
                      float* __restrict__ dout, float* __restrict__ din) {
  const int e = blockIdx.x * blockDim.x + threadIdx.x;
  if (e >= N_EDGES) return;
  atomicAdd(&dout[row[e]], 1.f);
  atomicAdd(&din[col[e]], 1.f);
}

// dst[col[e]] += (1/deg[sel]) * src[row[e], :]   (512 features, float4 per thread)
__global__ void k_prop(const float* __restrict__ src, float* __restrict__ dst,
                       const int* __restrict__ row, const int* __restrict__ col,
                       const float* __restrict__ deg, int use_row_deg) {
  const int e = blockIdx.x * 2 + (threadIdx.x >> 7);
  if (e >= N_EDGES) return;
  const int f = (threadIdx.x & 127) * 4;
  const int r = row[e], c = col[e];
  const float wgt = 1.0f / (use_row_deg ? deg[r] : deg[c]);
  const float4 v = *(const float4*)&src[(long)r * DIN + f];
  float* d = &dst[(long)c * DIN + f];
  atomicAdd(d + 0, wgt * v.x);
  atomicAdd(d + 1, wgt * v.y);
  atomicAdd(d + 2, wgt * v.z);
  atomicAdd(d + 3, wgt * v.w);
}

// Chebyshev recurrence: T2 = 2*T2 - X0
__global__ void k_cheb(float* __restrict__ t2, const float* __restrict__ x0) {
  const long i = (long)blockIdx.x * blockDim.x + threadIdx.x;
  if (i >= (long)N_NODES * DIN) return;
  t2[i] = 2.0f * t2[i] - x0[i];
}

// ---------------- weight stacking (WMMA-fragment swizzled) ----------------
// Logical stacked weight rows (grow 0..2559):
//   blk 0: W[0,0]+W[1,0]; blk 1: W[0,1]; blk 2: W[1,1]; blk 3: W[0,2]; blk 4: W[1,2]
__device__ inline float wstack_val(const float* __restrict__ W, int grow, int j) {
  const int blk = grow >> 9, r = grow & 511;
  if (blk == 0)
    return W[(long)r * EMB + j] + W[((long)3 * DIN + r) * EMB + j];
  const int dir = (blk - 1) & 1;
  const int hop = 1 + ((blk - 1) >> 1);
  return W[((long)(dir * 3 + hop) * DIN + r) * EMB + j];
}

// Swizzled layout: idx = ((cg*NKSTEP + kstep)*32 + lane)*8 + q
//   q&1 -> K row (grow = kstep*4 + 2*(lane>>4) + (q&1))
//   q>>1 -> column sub-tile; col = cg*64 + (q>>1)*16 + (lane&15)
// GEMM lane then loads its 8 B-fragment floats with two b128 loads (1KB/wave, coalesced).

__global__ void k_wswz_zr(const float* __restrict__ Wz, const float* __restrict__ Wr,
                          const float* __restrict__ bz, const float* __restrict__ br,
                          float* __restrict__ Wswz, float* __restrict__ bzr) {
  const int i = blockIdx.x * blockDim.x + threadIdx.x;   // 8*640*32*8 = 1,310,720
  if (i >= 8 * NKSTEP * 32 * 8) return;
  const int q = i & 7;
  const int lane = (i >> 3) & 31;
  const int t = i >> 8;
  const int kstep = t % NKSTEP;
  const int cg = t / NKSTEP;
  const int n = lane & 15;
  const int kb = (lane >> 4) * 2;
  const int grow = kstep * 4 + kb + (q & 1);
  const int col = cg * 64 + ((q >> 1) << 4) + n;
  Wswz[i] = (col < EMB) ? wstack_val(Wz, grow, col)
                        : wstack_val(Wr, grow, col - EMB);
  if (i < 2 * EMB) bzr[i] = (i < EMB) ? bz[i] : br[i - EMB];
}

__global__ void k_wswz_h(const float* __restrict__ Wh, float* __restrict__ Wswz) {
  const int i = blockIdx.x * blockDim.x + threadIdx.x;   // 4*640*32*8 = 655,360
  if (i >= 4 * NKSTEP * 32 * 8) return;
  const int q = i & 7;
  const int lane = (i >> 3) & 31;
  const int t = i >> 8;
  const int kstep = t % NKSTEP;
  const int cg = t / NKSTEP;
  const int n = lane & 15;
  const int kb = (lane >> 4) * 2;
  const int grow = kstep * 4 + kb + (q & 1);
  const int col = cg * 64 + ((q >> 1) << 4) + n;
  Wswz[i] = wstack_val(Wh, grow, col);
}

// ---------------- fp32 WMMA GEMM ----------------
// out[20000, ldout] = act( sum_blk basis_blk[20000,512] @ Wstack[blk] + bias )
// One wave owns a 16x64 output tile; K-step 4 via V_WMMA_F32_16X16X4_F32.
// B fragments come pre-swizzled: two b128 loads per K-step.

__global__ void __launch_bounds__(32)
k_gemm(const float* __restrict__ a0, const float* __restrict__ a1,
       const float* __restrict__ a2, const float* __restrict__ a3,
       const float* __restrict__ a4,
       const float* __restrict__ Wswz,
       const float* __restrict__ bias,
       float* __restrict__ out, int ldout, int act) {
  const int lane = threadIdx.x;
  const int row0 = blockIdx.x * 16;
  const int col0 = blockIdx.y * 64;
  const int m = lane & 15;                 // A row within tile / B-D column
  const int kb = (lane >> 4) * 2;          // K sub-offset per half-wave
  v8f c0 = {}, c1 = {}, c2 = {}, c3 = {};
  const float* bases[5] = {a0, a1, a2, a3, a4};
  const float4* wp =
      (const float4*)(Wswz + ((long)blockIdx.y * NKSTEP * 32 + lane) * 8);
#pragma unroll
  for (int blk = 0; blk < 5; ++blk) {
    const float* A = bases[blk] + (long)(row0 + m) * DIN + kb;
    for (int ks = 0; ks < 128; ++ks) {
      const v2f a = *(const v2f*)A;        // b64, A fragment (2 K values)
      A += 4;
      const float4 p0 = wp[0];             // b128 x2: 8 B-fragment floats
      const float4 p1 = wp[1];
      wp += 64;                            // 32 lanes * 8 floats per K-step
      const v2f b0 = {p0.x, p0.y};
      const v2f b1 = {p0.z, p0.w};
      const v2f b2 = {p1.x, p1.y};
      const v2f b3 = {p1.z, p1.w};
      c0 = __builtin_amdgcn_wmma_f32_16x16x4_f32(false, a, false, b0, (short)0, c0, false, false);
      c1 = __builtin_amdgcn_wmma_f32_16x16x4_f32(false, a, false, b1, (short)0, c1, false, false);
      c2 = __builtin_amdgcn_wmma_f32_16x16x4_f32(false, a, false, b2, (short)0, c2, false, false);
      c3 = __builtin_amdgcn_wmma_f32_16x16x4_f32(false, a, false, b3, (short)0, c3, false, false);
    }
  }
  const int rowsel = (lane >> 4) * 8;      // C/D layout: VGPR r -> M = r + 8*(lane>>4)
  auto epi = [&](const v8f& c, int cg) {
    const int coln = col0 + cg * 16 + m;
    const float bv = bias[coln];
#pragma unroll
    for (int r = 0; r < 8; ++r) {
      float val = c[r] + bv;
      val = (act == 0) ? (1.0f / (1.0f + expf(-val))) : tanhf(val);
      out[(long)(row0 + rowsel + r) * ldout + coln] = val;
    }
  };
  epi(c0, 0); epi(c1, 1); epi(c2, 2); epi(c3, 3);
}

// ---------------- GRU glue ----------------

__global__ void k_xrh(const float* __restrict__ ZR, const float* __restrict__ H0,
                      float* __restrict__ XRH) {
  const long i = (long)blockIdx.x * blockDim.x + threadIdx.x;
  if (i >= (long)N_NODES * EMB) return;
  const int n = (int)(i / EMB), j = (int)(i % EMB);
  XRH[(long)n * DIN + EMB + j] = ZR[(long)n * DIN + EMB + j] * H0[i];
}

__global__ void k_final(const float* __restrict__ ZR, const float* __restrict__ Ht,
                        const float* __restrict__ H0, float* __restrict__ out) {
  const long i = (long)blockIdx.x * blockDim.x + threadIdx.x;
  if (i >= (long)N_NODES * EMB) return;
  const int n = (int)(i / EMB), j = (int)(i % EMB);
  const float z = ZR[(long)n * DIN + j];
  const float h = z * H0[i] + (1.0f - z) * Ht[i];
  out[i] = fmaxf(h, 0.f);
}

// ---------------- host ----------------

extern "C" void kernel_launch(void* const* d_in, const int* in_sizes, int n_in,
                              void* d_out, int out_size, void* d_ws, size_t ws_size,
                              hipStream_t stream) {
  const float* x      = (const float*)d_in[0];
  const int*   ei     = (const int*)d_in[1];
  const float* H0     = (const float*)d_in[2];
  const float* c1w    = (const float*)d_in[3];
  const float* c1b    = (const float*)d_in[4];
  const float* g1     = (const float*)d_in[5];
  const float* be1    = (const float*)d_in[6];
  const float* c2w    = (const float*)d_in[7];
  const float* c2b    = (const float*)d_in[8];
  const float* g2     = (const float*)d_in[9];
  const float* be2    = (const float*)d_in[10];
  const float* Wz     = (const float*)d_in[11];
  const float* bz     = (const float*)d_in[12];
  const float* Wr     = (const float*)d_in[13];
  const float* br     = (const float*)d_in[14];
  const float* Wh     = (const float*)d_in[15];
  const float* bh     = (const float*)d_in[16];
  const int* row = ei;
  const int* col = ei + N_EDGES;

  float* ws = (float*)d_ws;
  float* y1     = ws;                          // 19,840,000
  float* y2     = y1  + 19840000;              //  5,120,000
  float* stats1 = y2  + 5120000;               // 64
  float* stats2 = stats1 + 64;                 // 64
  float* degO   = stats2 + 64;                 // 20000
  float* degI   = degO + 20000;                // 20000
  float* XH     = degI + 20000;                // 10,240,000
  float* XRH    = XH   + 10240000;             // 10,240,000
  float* T1o    = XRH  + 10240000;             // 4x 10,240,000 contiguous
  float* T1i    = T1o  + 10240000;
  float* T2o    = T1i  + 10240000;
  float* T2i    = T2o  + 10240000;
  float* Wzr    = T2i  + 10240000;             // 1,310,720 (swizzled)
  float* bzr    = Wzr  + 1310720;              // 512
  float* Wsth   = bzr  + 512;                  // 655,360 (swizzled)
  float* ZR     = Wsth + 655360;               // 10,240,000
  float* Htl    = ZR   + 10240000;             // 5,120,000
  float* out    = (float*)d_out;

  // zero stats + degrees (contiguous)
  hipMemsetAsync(stats1, 0, (64 + 64 + 20000 + 20000) * sizeof(float), stream);

  // weight stacks, pre-swizzled into WMMA B-fragment order
  k_wswz_zr<<<(1310720 + 255) / 256, 256, 0, stream>>>(Wz, Wr, bz, br, Wzr, bzr);
  k_wswz_h <<<(655360 + 255) / 256, 256, 0, stream>>>(Wh, Wsth);

  // CNN encoder
  k_conv1<<<N_NODES, 256, 0, stream>>>(x, c1w, c1b, y1, stats1);
  k_bn1  <<<(19840000 + 255) / 256, 256, 0, stream>>>(y1, stats1, g1, be1);
  k_conv2<<<N_NODES, 256, 0, stream>>>(y1, c2w, c2b, y2, stats2);
  k_bn2  <<<(5120000 + 255) / 256, 256, 0, stream>>>(y2, stats2, g2, be2, H0, XH, XRH);

  // degrees
  k_deg<<<(N_EDGES + 255) / 256, 256, 0, stream>>>(row, col, degO, degI);

  // ---- Z/R phase: diffusion basis on XH ----
  hipMemsetAsync(T1o, 0, 4ull * 10240000 * sizeof(float), stream);
  k_prop<<<N_EDGES / 2, 256, 0, stream>>>(XH, T1o, row, col, degO, 1);
  k_prop<<<N_EDGES / 2, 256, 0, stream>>>(XH, T1i, row, col, degI, 0);
  k_prop<<<N_EDGES / 2, 256, 0, stream>>>(T1o, T2o, row, col, degO, 1);
  k_prop<<<N_EDGES / 2, 256, 0, stream>>>(T1i, T2i, row, col, degI, 0);
  k_cheb<<<(10240000 + 255) / 256, 256, 0, stream>>>(T2o, XH);
  k_cheb<<<(10240000 + 255) / 256, 256, 0, stream>>>(T2i, XH);

  // fused Z|R GEMM: [20000,2560] x [2560,512], sigmoid epilogue
  {
    dim3 g(N_NODES / 16, (2 * EMB) / 64);
    k_gemm<<<g, 32, 0, stream>>>(XH, T1o, T1i, T2o, T2i, Wzr, bzr, ZR, 2 * EMB, 0);
  }

  // XRH second half = R * H0
  k_xrh<<<(5120000 + 255) / 256, 256, 0, stream>>>(ZR, H0, XRH);

  // ---- H~ phase: diffusion basis on XRH ----
  hipMemsetAsync(T1o, 0, 4ull * 10240000 * sizeof(float), stream);
  k_prop<<<N_EDGES / 2, 256, 0, stream>>>(XRH, T1o, row, col, degO, 1);
  k_prop<<<N_EDGES / 2, 256, 0, stream>>>(XRH, T1i, row, col, degI, 0);
  k_prop<<<N_EDGES / 2, 256, 0, stream>>>(T1o, T2o, row, col, degO, 1);
  k_prop<<<N_EDGES / 2, 256, 0, stream>>>(T1i, T2i, row, col, degI, 0);
  k_cheb<<<(10240000 + 255) / 256, 256, 0, stream>>>(T2o, XRH);
  k_cheb<<<(10240000 + 255) / 256, 256, 0, stream>>>(T2i, XRH);

  // H~ GEMM: [20000,2560] x [2560,256], tanh epilogue
  {
    dim3 g(N_NODES / 16, EMB / 64);
    k_gemm<<<g, 32, 0, stream>>>(XRH, T1o, T1i, T2o, T2i, Wsth, bh, Htl, EMB, 1);
  }

  // H = relu(Z*H0 + (1-Z)*H~)
  k_final<<<(5120000 + 255) / 256, 256, 0, stream>>>(ZR, Htl, H0, out);
}